// DFDgraph_31044023616140
// MI455X (gfx1250) — compile-verified
//
#include <hip/hip_runtime.h>
#include <math.h>

typedef float v2f __attribute__((ext_vector_type(2)));
typedef float v8f __attribute__((ext_vector_type(8)));

#define BB    8
#define NN    2048
#define TT    288
#define FBINS 145      // T/2+1
#define HH    64
#define EMB   24
#define HCAT  88       // H + EMB
#define KTOP  10
#define TILE  16
#define CHUNK 256
#define CPAD  (CHUNK + 4)   // LDS row pad: bank-conflict-free
#define NCHUNK (NN / CHUNK)

// ---------------------------------------------------------------------------
// Kernel 1: per-row feature pipeline.
// DFT(288)->|.|, min_max + l2, [xn@Wd0 | te] @ We0, relu, LayerNorm.
// Emits hA = h * W (bilinear-scaled) and hB = h into workspace.
// ---------------------------------------------------------------------------
__global__ __launch_bounds__(160)
void feat_kernel(const float* __restrict__ x, const float* __restrict__ temb,
                 const float* __restrict__ Wd0, const float* __restrict__ We0,
                 const float* __restrict__ Wv,
                 float* __restrict__ hA, float* __restrict__ hB)
{
  __shared__ float xs[TT];
  __shared__ float twc[TT];
  __shared__ float tws[TT];
  __shared__ float bins[FBINS];
  __shared__ float hcat[HCAT];
  __shared__ float hh[HH];
  __shared__ float s_mn, s_mx, s_inv, s_mean, s_istd;

  const int row = blockIdx.x;          // flattened b*N + n  (0..16383)
  const int tid = threadIdx.x;

  // load x row + build twiddle table exp(-2*pi*i*m/288)
  const float* xrow = x + (size_t)row * TT;
  for (int t = tid; t < TT; t += 160) xs[t] = xrow[t];
  const float twopi = 6.28318530717958647692f;
  for (int m2 = tid; m2 < TT; m2 += 160) {
    float sv, cv;
    __sincosf(-twopi * (float)m2 / (float)TT, &sv, &cv);
    twc[m2] = cv; tws[m2] = sv;
  }
  // t_emb branch: 24 values, serial in one thread, into hcat[64..87]
  if (tid == 32) {
    const float* te = temb + (size_t)row * EMB;
    float v[EMB];
    float mn = 1e30f, mx = -1e30f;
    for (int i = 0; i < EMB; ++i) { v[i] = te[i]; mn = fminf(mn, v[i]); mx = fmaxf(mx, v[i]); }
    const float den = 1.0f / (mx - mn + 1.0f);
    float ss = 0.f;
    for (int i = 0; i < EMB; ++i) { v[i] = (v[i] - mn) * den; ss += v[i] * v[i]; }
    const float inv = 1.0f / fmaxf(sqrtf(ss), 1e-12f);
    for (int i = 0; i < EMB; ++i) hcat[HH + i] = v[i] * inv;
  }
  __syncthreads();

  // naive DFT, idx = (k*t) mod 288 kept incrementally (pure LDS inner loop)
  if (tid < FBINS) {
    float re = 0.f, im = 0.f;
    int idx = 0;
    for (int t = 0; t < TT; ++t) {
      const float xv = xs[t];
      re = fmaf(xv, twc[idx], re);
      im = fmaf(xv, tws[idx], im);
      idx += tid; if (idx >= TT) idx -= TT;
    }
    bins[tid] = sqrtf(re * re + im * im) * 0.05892556509887896f;  // * 1/sqrt(288)
  }
  __syncthreads();
  if (tid == 0) {
    float mn = 1e30f, mx = -1e30f;
    for (int i = 0; i < FBINS; ++i) { mn = fminf(mn, bins[i]); mx = fmaxf(mx, bins[i]); }
    s_mn = mn; s_mx = mx;
  }
  __syncthreads();
  if (tid < FBINS) bins[tid] = (bins[tid] - s_mn) / (s_mx - s_mn + 1.0f);
  __syncthreads();
  if (tid == 0) {
    float ss = 0.f;
    for (int i = 0; i < FBINS; ++i) ss += bins[i] * bins[i];
    s_inv = 1.0f / fmaxf(sqrtf(ss), 1e-12f);
  }
  __syncthreads();
  if (tid < FBINS) bins[tid] *= s_inv;
  __syncthreads();

  // xn @ Wd0  -> hcat[0..63]   (Wd0 is (145,64) row-major, coalesced over tid)
  if (tid < HH) {
    float acc = 0.f;
    for (int i = 0; i < FBINS; ++i) acc = fmaf(bins[i], Wd0[i * HH + tid], acc);
    hcat[tid] = acc;
  }
  __syncthreads();
  // relu(hcat @ We0)  (We0 is (88,64) row-major)
  if (tid < HH) {
    float acc = 0.f;
    for (int i = 0; i < HCAT; ++i) acc = fmaf(hcat[i], We0[i * HH + tid], acc);
    hh[tid] = fmaxf(acc, 0.f);
  }
  __syncthreads();
  if (tid == 0) {
    float mu = 0.f;
    for (int i = 0; i < HH; ++i) mu += hh[i];
    mu *= (1.0f / HH);
    float var = 0.f;
    for (int i = 0; i < HH; ++i) { const float d = hh[i] - mu; var = fmaf(d, d, var); }
    var *= (1.0f / HH);
    s_mean = mu; s_istd = rsqrtf(var + 1e-8f);
  }
  __syncthreads();
  if (tid < HH) {
    const float v = (hh[tid] - s_mean) * s_istd;
    hB[(size_t)row * HH + tid] = v;
    hA[(size_t)row * HH + tid] = v * Wv[tid];
  }
}

// ---------------------------------------------------------------------------
// Kernel 2: adj = relu(hA @ hB^T) per batch via V_WMMA_F32_16X16X4_F32,
// software-pipelined with a dedicated top-K wave:
//   waves 0-7 : WMMA GEMM of chunk i into ping-pong LDS buffer
//   wave  8   : streaming top-10 merge of chunk i-1 (state held in registers)
// adj never touches HBM; final phase writes coalesced zero+scatter rows.
// ---------------------------------------------------------------------------
__global__ __launch_bounds__(288)
void adj_topk_kernel(const float* __restrict__ hA, const float* __restrict__ hB,
                     float* __restrict__ z)
{
  __shared__ float adjLds[2][TILE][CPAD];   // 2 x ~16.6 KB ping-pong
  __shared__ float topV[TILE][KTOP];        // normalized values (written once)
  __shared__ int   topI[TILE][KTOP];

  const int tid  = threadIdx.x;
  const int lane = tid & 31;
  const int wave = tid >> 5;              // 0..8 (9 waves)
  const int rowBase  = blockIdx.x * TILE; // flattened row tile (never crosses batch)
  const int jRowBase = (rowBase / NN) * NN;

  // A fragment preload (compute waves): 16 k-steps of v2f per lane
  const int m  = lane & 15;               // M row (A) / N col (B)
  const int hi = lane >> 4;               // selects K pair {0,1} vs {2,3}
  float a0[16], a1[16];
  if (wave < 8) {
    const float* Arow = hA + (size_t)(rowBase + m) * HH;
#pragma unroll
    for (int s = 0; s < 16; ++s) { a0[s] = Arow[4 * s + 2 * hi]; a1[s] = Arow[4 * s + 2 * hi + 1]; }
  }

  // merge wave register state: per-lane (== per-row) sorted top-10
  float mtv[KTOP]; int mti[KTOP];
#pragma unroll
  for (int q = 0; q < KTOP; ++q) { mtv[q] = -1.0f; mti[q] = 0; }

  for (int it = 0; it <= NCHUNK; ++it) {
    if (wave < 8 && it < NCHUNK) {
      const int jChunkBase = it * CHUNK;
      float (*buf)[CPAD] = adjLds[it & 1];
#pragma unroll
      for (int w = 0; w < 2; ++w) {
        const int jt    = wave + 8 * w;        // 16 j-tiles per chunk over 8 waves
        const int jbase = jChunkBase + jt * TILE;
        const float* Brow = hB + (size_t)(jRowBase + jbase + m) * HH;
        // stage all 16 B fragments first -> one load clause + one wait,
        // then an uninterrupted 16-deep WMMA accumulation chain
        float b0[16], b1[16];
#pragma unroll
        for (int s = 0; s < 16; ++s) { b0[s] = Brow[4 * s + 2 * hi]; b1[s] = Brow[4 * s + 2 * hi + 1]; }
        v8f c = {};
#pragma unroll
        for (int s = 0; s < 16; ++s) {         // K = 64 in steps of 4
          v2f av; av.x = a0[s]; av.y = a1[s];
          v2f bv; bv.x = b0[s]; bv.y = b1[s];
          c = __builtin_amdgcn_wmma_f32_16x16x4_f32(false, av, false, bv,
                                                    (short)0, c, false, false);
        }
#pragma unroll
        for (int r = 0; r < 8; ++r) {          // C layout: row = r + 8*hi, col = m
          const float v = c[r];
          buf[r + 8 * hi][jt * TILE + m] = v > 0.f ? v : 0.f;
        }
      }
    }
    if (wave == 8 && lane < TILE && it > 0) {
      // merge previous chunk; strict '>' keeps lowest index on ties (jax top_k)
      const int jcb = (it - 1) * CHUNK;
      const float* rowp = adjLds[(it - 1) & 1][lane];
      for (int j = 0; j < CHUNK; ++j) {
        const float v = rowp[j];
        if (v > mtv[KTOP - 1]) {
          int p = KTOP - 1;
          while (p > 0 && v > mtv[p - 1]) { mtv[p] = mtv[p - 1]; mti[p] = mti[p - 1]; --p; }
          mtv[p] = v; mti[p] = jcb + j;
        }
      }
    }
    __syncthreads();
  }

  // merge wave publishes normalized top-10 (z = vals / (sum + 1e-5))
  if (wave == 8 && lane < TILE) {
    float s = 0.f;
#pragma unroll
    for (int q = 0; q < KTOP; ++q) s += mtv[q];
    const float sc = 1.0f / (s + 1e-5f);
#pragma unroll
    for (int q = 0; q < KTOP; ++q) { topV[lane][q] = mtv[q] * sc; topI[lane][q] = mti[q]; }
  }
  __syncthreads();

  // write full output rows: zeros with the 10 normalized values substituted.
  // consecutive lanes of a half-wave write consecutive float4s (coalesced).
  if (tid < 256) {
    const int r  = tid >> 4;                  // row within tile
    const int cl = (tid & 15) * 4;            // interleaved column phase
    float tv[KTOP]; int ti[KTOP];
#pragma unroll
    for (int q = 0; q < KTOP; ++q) { tv[q] = topV[r][q]; ti[q] = topI[r][q]; }
    float* outRow = z + (size_t)(rowBase + r) * NN;
    for (int it2 = 0; it2 < NN / 64; ++it2) { // 32 iterations of 64 cols/row-group
      const int c = it2 * 64 + cl;
      float4 v = make_float4(0.f, 0.f, 0.f, 0.f);
#pragma unroll
      for (int q = 0; q < KTOP; ++q) {
        const int d = ti[q] - c;
        if (d >= 0 && d < 4) (&v.x)[d] = tv[q];
      }
      *(float4*)(outRow + c) = v;
    }
  }
}

extern "C" void kernel_launch(void* const* d_in, const int* in_sizes, int n_in,
                              void* d_out, int out_size, void* d_ws, size_t ws_size,
                              hipStream_t stream) {
  (void)in_sizes; (void)n_in; (void)out_size; (void)ws_size;
  const float* x    = (const float*)d_in[0];
  const float* temb = (const float*)d_in[1];
  const float* Wd0  = (const float*)d_in[2];
  const float* We0  = (const float*)d_in[3];
  const float* Wv   = (const float*)d_in[4];
  float* z  = (float*)d_out;
  float* hA = (float*)d_ws;                         // 16384*64 f32 = 4 MB
  float* hB = hA + (size_t)BB * NN * HH;            // next 4 MB

  feat_kernel<<<BB * NN, 160, 0, stream>>>(x, temb, Wd0, We0, Wv, hA, hB);
  adj_topk_kernel<<<(BB * NN) / TILE, 288, 0, stream>>>(hA, hB, z);
}